// BuildingModule_38079180046971
// MI455X (gfx1250) — compile-verified
//
#include <hip/hip_runtime.h>
#include <stdint.h>

// B=8192 independent 3-state trajectories, T=1024 steps. Sequential in t,
// parallel in batch. Memory floor ~352MB => ~15us at 23.3 TB/s; with only
// 256 waves of parallelism the per-step serial chain (rcp -> fma tree ->
// exp) is the limiter, so:
//   * u[] staged via gfx1250 async global->LDS pipeline (ASYNCcnt),
//     double-buffered, 8KB/chunk/wave, fully coalesced 16B/lane issues.
//   * reciprocals: raw v_rcp_f32 (1 ulp).
//   * exponentials: v_exp_f32 directly, with H/C * log2(e) prefolded
//     (removes 3 dependent v_mul per step from the critical path).
// No matrix contraction exists in this workload => WMMA not applicable.

#define WAVES_PER_BLOCK 2
#define CHUNK 8                    // timesteps staged per chunk
#define ROW_DW (CHUNK * 8 + 4)     // 64 data dwords + 4 pad = 68 (272B, 16B aligned)
#define BUF_DW (32 * ROW_DW)       // one buffer for one wave (32 batch rows)
#define ISSUES_PER_CHUNK 16        // 32 rows * 256B / (32 lanes * 16B)

struct __align__(4) f3 { float x, y, z; };

__device__ __forceinline__ void async_b128_to_lds(uint32_t lds_addr, const float* gaddr) {
    // GLOBAL_LOAD_ASYNC_TO_LDS_B128, GV mode: per-lane 16B global -> LDS.
    // Tracked by ASYNCcnt; completes in order with other async loads.
    asm volatile("global_load_async_to_lds_b128 %0, %1, off"
                 :: "v"(lds_addr), "v"(gaddr)
                 : "memory");
}

__global__ __launch_bounds__(WAVES_PER_BLOCK * 32, 1)
void building_scan_kernel(const float* __restrict__ x0,
                          const float* __restrict__ u,
                          const float* __restrict__ lam,
                          float* __restrict__ out,
                          int B, int T)
{
    __shared__ __align__(16) float lds[WAVES_PER_BLOCK][2][BUF_DW];

    const int wave = threadIdx.x >> 5;
    const int lane = threadIdx.x & 31;
    const int b0   = (blockIdx.x * WAVES_PER_BLOCK + wave) * 32; // wave's first batch elem
    const int b    = b0 + lane;                                  // this lane's batch elem

    // --- async staging addresses for this lane's role in cooperative loads ---
    const int rsel = lane >> 4;   // which of the 2 rows this lane serves per issue
    const int c16  = lane & 15;   // 16-byte slice within a 256B row
    const uint32_t lds_base = (uint32_t)(uintptr_t)&lds[wave][0][0]; // low 32b of
    // a generic shared pointer == wave-relative LDS byte address (HW adds LDS_BASE).

    const int nchunk = T / CHUNK;

    // Issue one chunk (8 timesteps x 32 batch rows = 8KB) into buffer bsel.
    auto issue_chunk = [&](int k, int bsel) {
        const float* gbase = u + ((size_t)b0 * T + (size_t)k * CHUNK) * 8;
        const uint32_t lbase = lds_base + (uint32_t)(bsel * BUF_DW * 4);
#pragma unroll
        for (int pair = 0; pair < ISSUES_PER_CHUNK; ++pair) {
            const int row = pair * 2 + rsel;
            const float* g = gbase + (size_t)row * T * 8 + c16 * 4;  // 256B/row, coalesced
            const uint32_t l = lbase + (uint32_t)(row * ROW_DW * 4 + c16 * 16);
            async_b128_to_lds(l, g);
        }
    };

    // Prologue: get two chunks in flight before touching anything else.
    issue_chunk(0, 0);
    issue_chunk(1, 1);

    // Per-thread constants while the loads fly.
    const float e12 = __expf(lam[0]);
    const float e23 = __expf(lam[1]);
    const float ee0 = __expf(lam[2]),  ee1 = __expf(lam[3]),  ee2 = __expf(lam[4]);
    const float es0 = __expf(lam[5]),  es1 = __expf(lam[6]),  es2 = __expf(lam[7]);
    const float eh0 = __expf(lam[8]),  eh1 = __expf(lam[9]),  eh2 = __expf(lam[10]);
    const float ec0 = __expf(lam[11]), ec1 = __expf(lam[12]), ec2 = __expf(lam[13]);
    // H/C[k] * log2(e), folded in double precision, rounded once to f32:
    // exp(H*s/C) == exp2(HL_k * s), evaluated by a single v_exp_f32.
    const float HL0 = (float)(60.0 / 10665991.0 * 1.4426950408889634);
    const float HL1 = (float)(60.0 / 27000000.0 * 1.4426950408889634);
    const float HL2 = (float)(60.0 /  7953253.0 * 1.4426950408889634);

    float X0 = x0[b * 3 + 0];
    float X1 = x0[b * 3 + 1];
    float X2 = x0[b * 3 + 2];

    const size_t ob = (size_t)b * T * 3;
    {   // t = 0: x0 verbatim
        f3 v = {X0, X1, X2};
        *reinterpret_cast<f3*>(out + ob) = v;
    }

    for (int k = 0; k < nchunk; ++k) {
        // Ensure chunk k resident; keep chunk k+1 (16 issues) in flight if it exists.
        if (k + 1 < nchunk) asm volatile("s_wait_asynccnt 16" ::: "memory");
        else                asm volatile("s_wait_asynccnt 0"  ::: "memory");

        const int bsel = k & 1;
#pragma unroll
        for (int i = 0; i < CHUNK; ++i) {
            const int t = k * CHUNK + i;
            const float4* rp = reinterpret_cast<const float4*>(
                &lds[wave][bsel][lane * ROW_DW + i * 8]);
            const float4 ua = rp[0];   // u0 u1 u2 u3
            const float4 ub = rp[1];   // u4 u5 u6 u7
            const float u0v = ua.x;

            if (t > 0) {
                // output index t = state after step t-1, masked by u[b,t,0]
                const bool bad = (u0v < 1e-6f);
                f3 v;
                v.x = bad ? -1.0f : X0;
                v.y = bad ? -1.0f : X1;
                v.z = bad ? -1.0f : X2;
                *reinterpret_cast<f3*>(out + ob + (size_t)t * 3) = v;
            }
            if (t < T - 1) {
                // raw v_rcp_f32: ~1 ulp, shortens the serial chain vs IEEE div
                const float i0 = __builtin_amdgcn_rcpf(X0);
                const float i1 = __builtin_amdgcn_rcpf(X1);
                const float i2 = __builtin_amdgcn_rcpf(X2);
                const float p12 = e12 * (i0 - i1);
                const float p23 = e23 * (i1 - i2);
                const float ax0 =  p12 * X1;
                const float ax1 =  p23 * X2 - p12 * X0;
                const float ax2 = -p23 * X1;
                // ee*(inv - 1/u0)*u0 == ee*(u0*inv - 1)
                const float bu0 = ee0 * (u0v * i0 - 1.0f) + es0 * ua.y + eh0 * ua.z + ec0 * ub.y;
                const float bu1 = ee1 * (u0v * i1 - 1.0f) + es1 * ua.y + eh1 * ua.w + ec1 * ub.z;
                const float bu2 = ee2 * (u0v * i2 - 1.0f) + es2 * ua.y + eh2 * ub.x + ec2 * ub.w;
                // single v_exp_f32 each (base-2 with prefolded constant)
                X0 *= __builtin_amdgcn_exp2f(HL0 * (ax0 + bu0));
                X1 *= __builtin_amdgcn_exp2f(HL1 * (ax1 + bu1));
                X2 *= __builtin_amdgcn_exp2f(HL2 * (ax2 + bu2));
            }
        }
        // All LDS reads of buffer bsel have been consumed (DScnt waits precede the
        // value uses above) => safe to recycle it for chunk k+2.
        if (k + 2 < nchunk) issue_chunk(k + 2, bsel);
    }
}

extern "C" void kernel_launch(void* const* d_in, const int* in_sizes, int n_in,
                              void* d_out, int out_size, void* d_ws, size_t ws_size,
                              hipStream_t stream) {
    const float* x0  = (const float*)d_in[0];   // (B, 3)
    const float* u   = (const float*)d_in[1];   // (B, T, 8)
    const float* lam = (const float*)d_in[2];   // (14,)
    float* out = (float*)d_out;                 // (B, T, 3)

    const int B = in_sizes[0] / 3;                       // 8192
    const int T = (int)((long long)in_sizes[1] / ((long long)B * 8)); // 1024

    const int threads = WAVES_PER_BLOCK * 32;            // 64
    const int blocks  = B / threads;                     // 128
    building_scan_kernel<<<blocks, threads, 0, stream>>>(x0, u, lam, out, B, T);
}